// DGCNN_54073638257098
// MI455X (gfx1250) — compile-verified
//
#include <hip/hip_runtime.h>

#define KNN 20
#define NPTS 2048
#define NB 8
#define PTOT (NB*NPTS)
#define GGRID 1024
#define NEG_INF -3.402823466e38f
#define POS_INF  3.402823466e38f

typedef float v2f __attribute__((ext_vector_type(2)));
typedef float v8f __attribute__((ext_vector_type(8)));

// ---------------------------------------------------------------------------
// C[M,Nm] = A[M,Kd](row-major, lda) * B(packed K-pair-major) (+ bias[n]).
// B packing: element (k, n) lives at Bt[((k>>1)*ldb + n)*2 + (k&1)] so each
// lane's WMMA B fragment (k0, k0+1) at column n is ONE contiguous b64 load.
// Requirements: Kd % 4 == 0 (callers zero-pad), lda even, M % 32 == 0,
// Nm % 64 == 0. One wave computes a 32(M) x 64(N) tile; grid=(Nm/64, M/32).
// Fragment layouts per CDNA5 ISA (v_wmma_f32_16x16x4_f32):
//   A: lane l -> M = l%16 ; vgpr {0,1} hold K = kk + (l<16?0:2) + {0,1}
//   B: lane l -> N = l%16 ; vgpr {0,1} hold K = kk + (l<16?0:2) + {0,1}
//   D: acc[r], lane l -> M = r + (l<16?0:8), N = l%16
// ---------------------------------------------------------------------------
__global__ void __launch_bounds__(32)
gemm_wmma_f32(const float* __restrict__ A, int lda,
              const float* __restrict__ Bt, int ldb,
              float* __restrict__ C, int ldc,
              int Kd, const float* __restrict__ bias)
{
  const int lane = threadIdx.x;
  const int row0 = blockIdx.y << 5;
  const int col0 = blockIdx.x << 6;
  v8f a00 = {}; v8f a01 = {}; v8f a02 = {}; v8f a03 = {};
  v8f a10 = {}; v8f a11 = {}; v8f a12 = {}; v8f a13 = {};
  const int kh = (lane >> 4) << 1;              // 0 or 2 (even)
  const int nl = col0 + (lane & 15);
  const float* Ar0 = A + (size_t)(row0 + (lane & 15)) * lda;
  const float* Ar1 = Ar0 + (size_t)16 * lda;
  for (int kk = 0; kk < Kd; kk += 4) {
    const int k0 = kk + kh;
    const v2f va0 = *(const v2f*)(Ar0 + k0);
    const v2f va1 = *(const v2f*)(Ar1 + k0);
    const float* Bp = Bt + (((size_t)(k0 >> 1) * ldb + nl) << 1);
    const v2f b0 = *(const v2f*)(Bp);
    const v2f b1 = *(const v2f*)(Bp + 32);
    const v2f b2 = *(const v2f*)(Bp + 64);
    const v2f b3 = *(const v2f*)(Bp + 96);
    a00 = __builtin_amdgcn_wmma_f32_16x16x4_f32(false, va0, false, b0, (short)0, a00, false, false);
    a01 = __builtin_amdgcn_wmma_f32_16x16x4_f32(false, va0, false, b1, (short)0, a01, false, false);
    a02 = __builtin_amdgcn_wmma_f32_16x16x4_f32(false, va0, false, b2, (short)0, a02, false, false);
    a03 = __builtin_amdgcn_wmma_f32_16x16x4_f32(false, va0, false, b3, (short)0, a03, false, false);
    a10 = __builtin_amdgcn_wmma_f32_16x16x4_f32(false, va1, false, b0, (short)0, a10, false, false);
    a11 = __builtin_amdgcn_wmma_f32_16x16x4_f32(false, va1, false, b1, (short)0, a11, false, false);
    a12 = __builtin_amdgcn_wmma_f32_16x16x4_f32(false, va1, false, b2, (short)0, a12, false, false);
    a13 = __builtin_amdgcn_wmma_f32_16x16x4_f32(false, va1, false, b3, (short)0, a13, false, false);
  }
  const int rb = (lane >> 4) << 3;              // 0 or 8
  const int c0 = col0 + (lane & 15);
  const float bb0 = bias ? bias[c0]      : 0.f;
  const float bb1 = bias ? bias[c0 + 16] : 0.f;
  const float bb2 = bias ? bias[c0 + 32] : 0.f;
  const float bb3 = bias ? bias[c0 + 48] : 0.f;
#pragma unroll
  for (int r = 0; r < 8; ++r) {
    const size_t ro0 = (size_t)(row0 + rb + r) * ldc;
    const size_t ro1 = ro0 + (size_t)16 * ldc;
    C[ro0 + c0]      = a00[r] + bb0;
    C[ro0 + c0 + 16] = a01[r] + bb1;
    C[ro0 + c0 + 32] = a02[r] + bb2;
    C[ro0 + c0 + 48] = a03[r] + bb3;
    C[ro1 + c0]      = a10[r] + bb0;
    C[ro1 + c0 + 16] = a11[r] + bb1;
    C[ro1 + c0 + 32] = a12[r] + bb2;
    C[ro1 + c0 + 48] = a13[r] + bb3;
  }
}

// xpad[p*4 + c] = (c < 3) ? x[p*3 + c] : 0
__global__ void pad_x(const float* __restrict__ x, float* __restrict__ xpad)
{
  int i = blockIdx.x * blockDim.x + threadIdx.x;
  if (i >= PTOT * 4) return;
  int p = i >> 2;
  int c = i & 3;
  xpad[i] = (c < 3) ? x[p * 3 + c] : 0.f;
}

// Packed K-pair transpose: dst[((k>>1)*n + j)*2 + (k&1)] = (k<c)? src[j*lds+k] : 0
// for k in [0, kpad), j in [0, n). kpad % 4 == 0.
__global__ void transpose_pk(const float* __restrict__ src, int lds_,
                             float* __restrict__ dst, int n, int c, int kpad)
{
  int i = blockIdx.x * blockDim.x + threadIdx.x;
  if (i >= kpad * n) return;
  int k = i / n;
  int j = i - k * n;
  float v = (k < c) ? src[(size_t)j * lds_ + k] : 0.f;
  dst[(((size_t)(k >> 1) * n + j) << 1) + (k & 1)] = v;
}

// Packed weight prep: Bt1 <- W[:, :C]^T ; BtD <- (W[:, C:] - W[:, :C])^T
// zero-padded to kpad rows. Layout identical to transpose_pk's.
__global__ void make_bt_pk(const float* __restrict__ W, int Cin, int C, int O,
                           int kpad, float* __restrict__ Bt1, float* __restrict__ BtD)
{
  int i = blockIdx.x * blockDim.x + threadIdx.x;
  if (i >= kpad * O) return;
  int k = i / O;
  int o = i - k * O;
  float w1 = (k < C) ? W[(size_t)o * Cin + k]     : 0.f;
  float w2 = (k < C) ? W[(size_t)o * Cin + C + k] : 0.f;
  size_t d = (((size_t)(k >> 1) * O + o) << 1) + (k & 1);
  Bt1[d] = w1;
  BtD[d] = w2 - w1;
}

// x2[p] = sum_c h[p,c]^2
__global__ void rowsq(const float* __restrict__ h, int lda, int C,
                      float* __restrict__ x2)
{
  int p = blockIdx.x * blockDim.x + threadIdx.x;
  if (p >= PTOT) return;
  float s = 0.f;
  for (int c = 0; c < C; ++c) {
    float v = h[(size_t)p * lda + c];
    s += v * v;
  }
  x2[p] = s;
}

// Top-K by neg squared distance; ranking uses v_j = 2*G[n][j] - x2[j]
// (-x2[n] is a per-row constant shift). Writes GLOBAL row ids (b*N + j).
__global__ void topk_knn(const float* __restrict__ G, const float* __restrict__ x2b,
                         int* __restrict__ idxOut, int rowBase)
{
  int n = blockIdx.x * blockDim.x + threadIdx.x;
  if (n >= NPTS) return;
  const float* g = G + (size_t)n * NPTS;
  float vals[KNN]; int inds[KNN];
#pragma unroll
  for (int t = 0; t < KNN; ++t) { vals[t] = NEG_INF; inds[t] = 0; }
  for (int j = 0; j < NPTS; ++j) {
    float v = 2.0f * g[j] - x2b[j];
    if (v > vals[KNN - 1]) {
      float cv = v; int ci = j;
#pragma unroll
      for (int t = 0; t < KNN; ++t) {
        if (cv > vals[t]) {
          float tv = vals[t]; int ti = inds[t];
          vals[t] = cv; inds[t] = ci;
          cv = tv; ci = ti;
        }
      }
    }
  }
  int* o = idxOut + (size_t)(rowBase + n) * KNN;
#pragma unroll
  for (int t = 0; t < KNN; ++t) o[t] = rowBase + inds[t];
}

// Edge gather: v_k = A[idx[p][k]] + Cc[p]; emit max_k/min_k per (p,o) and
// deterministic per-block partial channel sums/sumsq. blockDim.x == O.
__global__ void edge_gather(const float* __restrict__ Amat, const float* __restrict__ Cc,
                            const int* __restrict__ idx,
                            float* __restrict__ mx, float* __restrict__ mn,
                            float* __restrict__ pS, float* __restrict__ pQ, int O)
{
  const int o = threadIdx.x;
  __shared__ int sIdx[KNN];
  float accS = 0.f, accQ = 0.f;
  for (int p = blockIdx.x; p < PTOT; p += gridDim.x) {
    if (threadIdx.x < KNN) sIdx[threadIdx.x] = idx[(size_t)p * KNN + threadIdx.x];
    __syncthreads();
    const float c = Cc[(size_t)p * O + o];
    float vmx = NEG_INF, vmn = POS_INF;
#pragma unroll
    for (int k = 0; k < KNN; ++k) {
      float a = Amat[(size_t)sIdx[k] * O + o];
      float v = a + c;
      vmx = fmaxf(vmx, v);
      vmn = fminf(vmn, v);
      accS += v;
      accQ += v * v;
    }
    mx[(size_t)p * O + o] = vmx;
    mn[(size_t)p * O + o] = vmn;
    __syncthreads();
  }
  pS[(size_t)blockIdx.x * O + o] = accS;
  pQ[(size_t)blockIdx.x * O + o] = accQ;
}

// Deterministic reduce of partials -> per-channel mean + BN scale
__global__ void bn_stats(const float* __restrict__ pS, const float* __restrict__ pQ,
                         const float* __restrict__ g,
                         float* __restrict__ chM, float* __restrict__ chS, int O)
{
  int o = threadIdx.x;
  if (o >= O) return;
  float s = 0.f, q = 0.f;
  for (int i = 0; i < GGRID; ++i) {
    s += pS[(size_t)i * O + o];
    q += pQ[(size_t)i * O + o];
  }
  const float cnt = (float)PTOT * (float)KNN;
  float m = s / cnt;
  float v = q / cnt - m * m;
  chM[o] = m;
  chS[o] = g[o] * rsqrtf(v + 1e-5f);
}

// max_k act(bn(h)) = act(bn(max_k h)) if scale>=0 else act(bn(min_k h))
__global__ void bn_act_write(const float* __restrict__ mx, const float* __restrict__ mn,
                             const float* __restrict__ chM, const float* __restrict__ chS,
                             const float* __restrict__ be,
                             float* __restrict__ xcat, int O, int outOff)
{
  int i = blockIdx.x * blockDim.x + threadIdx.x;
  if (i >= PTOT * O) return;
  int p = i / O;
  int o = i - p * O;
  float s = chS[o];
  float z = (s >= 0.f) ? mx[i] : mn[i];
  float y = s * (z - chM[o]) + be[o];
  y = (y >= 0.f) ? y : 0.2f * y;
  xcat[(size_t)p * 512 + outOff + o] = y;
}

// out[b,o] = max_n full[(b*N+n)*1024 + o] ; grid (1024/256, NB), block 256
__global__ void maxpool_n(const float* __restrict__ full, float* __restrict__ out)
{
  int o = blockIdx.x * 256 + threadIdx.x;
  int b = blockIdx.y;
  const float* base = full + (size_t)b * NPTS * 1024 + o;
  float m = NEG_INF;
  for (int n = 0; n < NPTS; ++n) m = fmaxf(m, base[(size_t)n * 1024]);
  out[b * 1024 + o] = m;
}

extern "C" void kernel_launch(void* const* d_in, const int* in_sizes, int n_in,
                              void* d_out, int out_size, void* d_ws, size_t ws_size,
                              hipStream_t stream)
{
  (void)in_sizes; (void)n_in; (void)out_size; (void)ws_size;
  const float* x = (const float*)d_in[0];
  const float* W[4]; const float* bI[4]; const float* gI[4]; const float* beI[4];
  for (int i = 0; i < 4; ++i) {
    W[i]   = (const float*)d_in[1 + 4 * i + 0];
    bI[i]  = (const float*)d_in[1 + 4 * i + 1];
    gI[i]  = (const float*)d_in[1 + 4 * i + 2];
    beI[i] = (const float*)d_in[1 + 4 * i + 3];
  }
  const float* Wf = (const float*)d_in[17];
  const float* bf = (const float*)d_in[18];

  float* ws = (float*)d_ws;
  size_t off = 0;
  float* xcat = ws;                 off += (size_t)PTOT * 512;   // layer outputs, concat
  float* G    = ws + off;           off += (size_t)NPTS * NPTS;  // per-batch Gram
  float* Amat = ws + off;           off += (size_t)PTOT * 256;   // h @ W1^T
  float* Cc   = ws + off;           off += (size_t)PTOT * 256;   // h @ (W2-W1)^T + b
  float* mxb  = ws + off;           off += (size_t)PTOT * 256;   // max over k
  float* mnb  = ws + off;           off += (size_t)PTOT * 256;   // min over k
  float* outFull = G;               // reuse G+Amat+Cc+mxb = 16384*1024 floats (dead by then)
  float* ht   = ws + off;           off += (size_t)128 * NPTS;   // per-batch h^T (packed)
  float* x2   = ws + off;           off += (size_t)PTOT;         // squared norms
  float* Bt1  = ws + off;           off += (size_t)128 * 256;    // W1^T packed
  float* BtD  = ws + off;           off += (size_t)128 * 256;    // (W2-W1)^T packed
  float* Btf  = ws + off;           off += (size_t)512 * 1024;   // Wf^T packed
  float* pS   = ws + off;           off += (size_t)GGRID * 256;  // partial chan sums
  float* pQ   = ws + off;           off += (size_t)GGRID * 256;  // partial chan sumsq
  float* chM  = ws + off;           off += 256;                  // channel mean
  float* chS  = ws + off;           off += 256;                  // channel scale
  float* xpad = ws + off;           off += (size_t)PTOT * 4;     // x zero-padded K=3->4
  int* idxw   = (int*)(ws + off);   off += (size_t)PTOT * KNN;   // global knn indices

  const int Cs[4]     = {3, 64, 64, 128};
  const int Os[4]     = {64, 64, 128, 256};
  const int inOff[4]  = {0, 0, 64, 128};
  const int outOff[4] = {0, 64, 128, 256};

  pad_x<<<(PTOT * 4 + 255) / 256, 256, 0, stream>>>(x, xpad);

  for (int l = 0; l < 4; ++l) {
    const int C = Cs[l], O = Os[l];
    const int Kp = (C + 3) & ~3;                 // 4, 64, 64, 128
    const float* hbase = (l == 0) ? xpad : (xcat + inOff[l]);
    const int ldh = (l == 0) ? 4 : 512;

    // ---- kNN: per-batch Gram matrix (WMMA f32) + top-20 scan ----
    rowsq<<<(PTOT + 255) / 256, 256, 0, stream>>>(hbase, ldh, Kp, x2);
    for (int b = 0; b < NB; ++b) {
      const float* hb = hbase + (size_t)b * NPTS * ldh;
      transpose_pk<<<(NPTS * Kp + 255) / 256, 256, 0, stream>>>(hb, ldh, ht, NPTS, Kp, Kp);
      gemm_wmma_f32<<<dim3(NPTS / 64, NPTS / 32), 32, 0, stream>>>(
          hb, ldh, ht, NPTS, G, NPTS, Kp, nullptr);
      topk_knn<<<NPTS / 64, 64, 0, stream>>>(G, x2 + b * NPTS, idxw, b * NPTS);
    }

    // ---- Edge conv as two dense GEMMs + gather ----
    make_bt_pk<<<(Kp * O + 255) / 256, 256, 0, stream>>>(W[l], 2 * C, C, O, Kp, Bt1, BtD);
    gemm_wmma_f32<<<dim3(O / 64, PTOT / 32), 32, 0, stream>>>(
        hbase, ldh, Bt1, O, Amat, O, Kp, nullptr);
    gemm_wmma_f32<<<dim3(O / 64, PTOT / 32), 32, 0, stream>>>(
        hbase, ldh, BtD, O, Cc, O, Kp, bI[l]);
    edge_gather<<<GGRID, O, 0, stream>>>(Amat, Cc, idxw, mxb, mnb, pS, pQ, O);
    bn_stats<<<1, 256, 0, stream>>>(pS, pQ, gI[l], chM, chS, O);
    bn_act_write<<<(PTOT * O + 255) / 256, 256, 0, stream>>>(
        mxb, mnb, chM, chS, beI[l], xcat, O, outOff[l]);
  }

  // ---- Final 1x1 conv (WMMA f32) + global max pool over N ----
  transpose_pk<<<(512 * 1024 + 255) / 256, 256, 0, stream>>>(Wf, 512, Btf, 1024, 512, 512);
  gemm_wmma_f32<<<dim3(1024 / 64, PTOT / 32), 32, 0, stream>>>(
      xcat, 512, Btf, 1024, outFull, 1024, 512, bf);
  maxpool_n<<<dim3(1024 / 256, NB), 256, 0, stream>>>(outFull, (float*)d_out);
}